// GATLayer_77704548319623
// MI455X (gfx1250) — compile-verified
//
#include <hip/hip_runtime.h>
#include <math.h>

// ---------------------------------------------------------------------------
// GAT layer, gfx1250 (MI455X, wave32, WMMA).
//
// Key identity: e[i,j] = s[i] + t[j] + b  =>  softmax over j is independent
// of i (row constants cancel), so attention[i,:] == softmax(t) for all i and
// h_prime[i,:] == softmax(t) @ Wh for all i.  Work collapses to:
//   (1) Wh = h @ W^T            -- 64 MFLOP, done with V_WMMA_F32_16X16X4_F32
//   (2) t = Wh @ a_right        -- fused into (1), in-register
//   (3) softmax(t) weighted sum -- 2 MB read, block partials + f32 atomics
//   (4) broadcast 64-float row  -- 2 MB write
// Total ~6 MB HBM traffic -> ~260 ns at 23.3 TB/s; f32 kept throughout to
// match the f32 reference.
// ---------------------------------------------------------------------------

typedef __attribute__((ext_vector_type(2))) float v2f;
typedef __attribute__((ext_vector_type(8))) float v8f;

#define NROWS 8192
#define DIM   64

// ---------------------------------------------------------------------------
// Kernel 1: per-wave 16-row tile of Wh = h @ W^T via WMMA f32 16x16x4,
// plus t = Wh @ a_right.
//
// A (16x4 f32, 2 VGPR/lane): lanes 0-15 hold M=lane, K={0,1}; lanes 16-31
// hold M=lane-16, K={2,3}  -> each lane loads a contiguous float2.
// B (4x16 f32, 2 VGPR/lane): lanes 0-15 hold N=lane, K={0,1}; lanes 16-31
// hold N=lane-16, K={2,3}. B[k][n] = W^T[k][n] = W[n][k] -> same contiguous
// float2 pattern on rows of W.
// C/D (16x16 f32, 8 VGPR/lane): VGPR r holds M=r (lanes 0-15, N=lane) and
// M=8+r (lanes 16-31, N=lane-16).
// ---------------------------------------------------------------------------
__global__ __launch_bounds__(256) void gat_wh_t_kernel(
    const float* __restrict__ h, const float* __restrict__ W,
    const float* __restrict__ attn_w, float* __restrict__ Wh,
    float* __restrict__ t) {
  const int lane = threadIdx.x & 31;
  const int wave = threadIdx.x >> 5;
  const int tile = blockIdx.x * 8 + wave;   // 512 tiles of 16 rows
  const int row0 = tile * 16;
  const int half = lane >> 4;               // 0: lanes 0-15, 1: lanes 16-31
  const int lrow = lane & 15;

  v8f acc0 = {}, acc1 = {}, acc2 = {}, acc3 = {};

  const float* hA = h + (size_t)(row0 + lrow) * DIM + half * 2;
  const float* wB = W + (size_t)lrow * DIM + half * 2;   // W row = output col

  #pragma unroll
  for (int kk = 0; kk < 16; ++kk) {         // K = 64 in steps of 4
    v2f a  = *(const v2f*)(hA + kk * 4);
    v2f b0 = *(const v2f*)(wB + 0 * 16 * DIM + kk * 4);
    v2f b1 = *(const v2f*)(wB + 1 * 16 * DIM + kk * 4);
    v2f b2 = *(const v2f*)(wB + 2 * 16 * DIM + kk * 4);
    v2f b3 = *(const v2f*)(wB + 3 * 16 * DIM + kk * 4);
    acc0 = __builtin_amdgcn_wmma_f32_16x16x4_f32(false, a, false, b0,
                                                 (short)0, acc0, false, false);
    acc1 = __builtin_amdgcn_wmma_f32_16x16x4_f32(false, a, false, b1,
                                                 (short)0, acc1, false, false);
    acc2 = __builtin_amdgcn_wmma_f32_16x16x4_f32(false, a, false, b2,
                                                 (short)0, acc2, false, false);
    acc3 = __builtin_amdgcn_wmma_f32_16x16x4_f32(false, a, false, b3,
                                                 (short)0, acc3, false, false);
  }

  // Store Wh tile (16x64). For fixed r, lanes 0-15 write one contiguous
  // 64B row segment, lanes 16-31 another.
  #pragma unroll
  for (int r = 0; r < 8; ++r) {
    const int m = half * 8 + r;
    float* rowp = Wh + (size_t)(row0 + m) * DIM + lrow;
    rowp[0]  = acc0[r];
    rowp[16] = acc1[r];
    rowp[32] = acc2[r];
    rowp[48] = acc3[r];
  }

  // t[row0+m] = sum_N Wh[m][N] * a_right[N]; a_right = attn_w[64..127].
  // (a_left and attn_b cancel in the row softmax and are never read.)
  const float aR0 = attn_w[64  + lrow];
  const float aR1 = attn_w[80  + lrow];
  const float aR2 = attn_w[96  + lrow];
  const float aR3 = attn_w[112 + lrow];
  #pragma unroll
  for (int r = 0; r < 8; ++r) {
    float pr = acc0[r] * aR0 + acc1[r] * aR1 + acc2[r] * aR2 + acc3[r] * aR3;
    pr += __shfl_xor(pr, 1, 32);   // butterfly over the 16 lanes of each half
    pr += __shfl_xor(pr, 2, 32);
    pr += __shfl_xor(pr, 4, 32);
    pr += __shfl_xor(pr, 8, 32);
    if (lrow == 0) t[row0 + half * 8 + r] = pr;
  }
}

// ---------------------------------------------------------------------------
// Kernel 2: tmax = max(t); zero the global accumulators (Z, wsum[64]).
// Must re-initialize every call: graph replay does not re-poison d_ws.
// ---------------------------------------------------------------------------
__global__ __launch_bounds__(1024) void gat_max_kernel(
    const float* __restrict__ t, float* __restrict__ tmax,
    float* __restrict__ Z, float* __restrict__ wsum) {
  __shared__ float s[1024];
  float m = -INFINITY;
  for (int i = threadIdx.x; i < NROWS; i += 1024) m = fmaxf(m, t[i]);
  s[threadIdx.x] = m;
  __syncthreads();
  for (int off = 512; off > 0; off >>= 1) {
    if (threadIdx.x < (unsigned)off)
      s[threadIdx.x] = fmaxf(s[threadIdx.x], s[threadIdx.x + off]);
    __syncthreads();
  }
  if (threadIdx.x == 0) { tmax[0] = s[0]; Z[0] = 0.0f; }
  if (threadIdx.x < 64) wsum[threadIdx.x] = 0.0f;
}

// ---------------------------------------------------------------------------
// Kernel 3: block-partial  Z += sum_j exp(t[j]-tmax),
//                          wsum[c] += sum_j exp(t[j]-tmax) * Wh[j][c]
// 64 blocks x 128 rows; threads = 4 row-groups x 64 columns (coalesced Wh).
// ---------------------------------------------------------------------------
__global__ __launch_bounds__(256) void gat_acc_kernel(
    const float* __restrict__ Wh, const float* __restrict__ t,
    const float* __restrict__ tmaxp, float* __restrict__ Z,
    float* __restrict__ wsum) {
  __shared__ float sc[4][64];
  __shared__ float sz[4];
  const int c  = threadIdx.x & 63;
  const int rg = threadIdx.x >> 6;
  const float tmax = tmaxp[0];
  const int j0 = blockIdx.x * 128;

  float accc = 0.0f, accz = 0.0f;
  for (int j = j0 + rg; j < j0 + 128; j += 4) {
    const float w = __expf(t[j] - tmax);
    accc += w * Wh[(size_t)j * DIM + c];
    accz += w;
  }
  sc[rg][c] = accc;
  if (c == 0) sz[rg] = accz;
  __syncthreads();
  if (rg == 0) {
    atomicAdd(&wsum[c], sc[0][c] + sc[1][c] + sc[2][c] + sc[3][c]);
    if (c == 0) atomicAdd(Z, sz[0] + sz[1] + sz[2] + sz[3]);
  }
}

// ---------------------------------------------------------------------------
// Kernel 4: every output row is identical: out[i][c] = wsum[c] / Z.
// float4 stores, 2 MB total.
// ---------------------------------------------------------------------------
__global__ __launch_bounds__(256) void gat_bcast_kernel(
    const float* __restrict__ wsum, const float* __restrict__ Z,
    float* __restrict__ out) {
  const int idx = blockIdx.x * 256 + threadIdx.x;     // float4 index
  const float invZ = 1.0f / Z[0];
  const int c4 = (idx & 15) * 4;                      // column group
  float4 v;
  v.x = wsum[c4 + 0] * invZ;
  v.y = wsum[c4 + 1] * invZ;
  v.z = wsum[c4 + 2] * invZ;
  v.w = wsum[c4 + 3] * invZ;
  ((float4*)out)[idx] = v;
}

// ---------------------------------------------------------------------------
// Workspace layout (floats):
//   [0, 524288)        Wh  (8192 x 64)
//   [524288, 532480)   t   (8192)
//   [532480]           tmax
//   [532481]           Z
//   [532482, 532546)   wsum[64]
// ---------------------------------------------------------------------------
extern "C" void kernel_launch(void* const* d_in, const int* in_sizes, int n_in,
                              void* d_out, int out_size, void* d_ws,
                              size_t ws_size, hipStream_t stream) {
  (void)in_sizes; (void)n_in; (void)out_size; (void)ws_size;
  const float* h      = (const float*)d_in[0];
  const float* W      = (const float*)d_in[1];
  const float* attn_w = (const float*)d_in[2];
  // d_in[3] (attn_b) cancels inside the row softmax and is not needed.

  float* ws   = (float*)d_ws;
  float* Wh   = ws;
  float* t    = ws + 524288;
  float* tmax = ws + 532480;
  float* Z    = ws + 532481;
  float* wsum = ws + 532482;

  gat_wh_t_kernel<<<64, 256, 0, stream>>>(h, W, attn_w, Wh, t);
  gat_max_kernel<<<1, 1024, 0, stream>>>(t, tmax, Z, wsum);
  gat_acc_kernel<<<64, 256, 0, stream>>>(Wh, t, tmax, Z, wsum);
  gat_bcast_kernel<<<(NROWS * DIM / 4) / 256, 256, 0, stream>>>(
      wsum, Z, (float*)d_out);
}